// IntraClassLoss_53137335386662
// MI455X (gfx1250) — compile-verified
//
#include <hip/hip_runtime.h>

typedef __attribute__((ext_vector_type(2))) float v2f;
typedef __attribute__((ext_vector_type(4))) float v4f;
typedef __attribute__((ext_vector_type(8))) float v8f;

#define CDIM   256          // feature channels
#define KCLS   16           // old classes
#define HWPIX  (128*128)    // pixels per image per channel
#define PIXB   2048         // pixel chunk per block (divides HWPIX)

// ---------------------------------------------------------------- kernel 0
__global__ __launch_bounds__(256) void k_zero(float* p, int n) {
    for (int i = blockIdx.x * 256 + threadIdx.x; i < n; i += gridDim.x * 256)
        p[i] = 0.0f;
}

// ---------------------------------------------------------------- kernel 1
// pseudo = argmax_c(outputs_old) * (labels < num_old); counts histogram
__global__ __launch_bounds__(256) void k_pseudo(const float* __restrict__ oo,
                                                const int*   __restrict__ labels,
                                                const int*   __restrict__ numoldp,
                                                unsigned char* __restrict__ seg,
                                                float* __restrict__ counts) {
    __shared__ float hist[KCLS];
    int t = threadIdx.x;
    if (t < KCLS) hist[t] = 0.0f;
    __syncthreads();

    int i  = blockIdx.x * 256 + t;              // pixel index, grid is exact
    int nb = *numoldp;
    int b  = i / HWPIX;
    int hw = i - b * HWPIX;
    const float* base = oo + (size_t)b * KCLS * HWPIX + hw;

    float best = base[0];
    int   arg  = 0;
#pragma unroll
    for (int c = 1; c < KCLS; ++c) {
        float v = base[(size_t)c * HWPIX];
        if (v > best) { best = v; arg = c; }    // strict > keeps first max (jnp.argmax)
    }
    int k = (labels[i] < nb) ? arg : 0;
    seg[i] = (unsigned char)k;
    atomicAdd(&hist[k], 1.0f);
    __syncthreads();
    if (t < KCLS) atomicAdd(&counts[t], hist[t]);
}

// ---------------------------------------------------------------- kernel 2
// One pass over f/fo. Each wave: 16-channel tile x PIXB pixels.
// Segment-sum via one-hot WMMA: D[16cls x 16ch] += A(onehot 16x4) * B(d 4x16).
// 8 pixels per macro-iteration via b128 loads; K->pixel mapping permuted so
// half-0 lanes own pixels {8it..+3}, half-1 lanes own {8it+4..+7} (A and B
// use the same mapping, so the contraction is unchanged).
__global__ __launch_bounds__(256) void k_main(const float* __restrict__ f,
                                              const float* __restrict__ fo,
                                              const unsigned char* __restrict__ seg,
                                              float* __restrict__ S,    // [KCLS*CDIM]
                                              float* __restrict__ Qg) { // [KCLS]
    __shared__ unsigned int segLds[PIXB / 4];

    const int chunk = blockIdx.x * PIXB;            // global pixel base (one batch image)
    // stage this block's seg bytes once (shared by all 8 waves)
    {
        const unsigned int* g = (const unsigned int*)(seg + chunk);
        for (int i = threadIdx.x; i < PIXB / 4; i += 256) segLds[i] = g[i];
    }
    __syncthreads();

    const int lane  = threadIdx.x & 31;
    const int wid   = threadIdx.x >> 5;             // 0..7
    const unsigned half = lane >> 4;                // 0: K0/K1, 1: K2/K3
    const int nIdx  = lane & 15;                    // B column index
    const int ctile = blockIdx.y * 8 + wid;         // 16 channel tiles total
    const int c     = ctile * 16 + nIdx;
    const unsigned mCls = (unsigned)nIdx;           // A-matrix row owned by this lane

    const int b   = chunk / HWPIX;
    const int hw0 = chunk - b * HWPIX;
    const float* fP  = f  + ((size_t)(b * CDIM + c)) * HWPIX + hw0;
    const float* foP = fo + ((size_t)(b * CDIM + c)) * HWPIX + hw0;

    v8f accS = {};
    v8f accQ = {};

    for (int it = 0; it < PIXB / 8; ++it) {
        const int p = it * 8 + 4 * half;            // this lane's 4 consecutive pixels
        // this lane's 4 seg labels, one ds_load_b32
        const unsigned sw = segLds[it * 2 + half];
        // one b128 load per tensor (16B aligned: p multiple of 4)
        v4f fv  = *(const v4f*)(fP  + p);
        v4f fov = *(const v4f*)(foP + p);
        v4f d  = fv - fov;
        v4f d2 = d * d;

        v2f a1, a2, b1, b2, q1, q2;
        a1.x = ((sw         & 0xffu) == mCls) ? 1.0f : 0.0f;   // pixels p, p+1
        a1.y = (((sw >> 8)  & 0xffu) == mCls) ? 1.0f : 0.0f;
        a2.x = (((sw >> 16) & 0xffu) == mCls) ? 1.0f : 0.0f;   // pixels p+2, p+3
        a2.y = ((sw >> 24)           == mCls) ? 1.0f : 0.0f;
        b1.x = d.x;  b1.y = d.y;   q1.x = d2.x; q1.y = d2.y;
        b2.x = d.z;  b2.y = d.w;   q2.x = d2.z; q2.y = d2.w;

        accS = __builtin_amdgcn_wmma_f32_16x16x4_f32(false, a1, false, b1,
                                                     (short)0, accS, false, false);
        accQ = __builtin_amdgcn_wmma_f32_16x16x4_f32(false, a1, false, q1,
                                                     (short)0, accQ, false, false);
        accS = __builtin_amdgcn_wmma_f32_16x16x4_f32(false, a2, false, b2,
                                                     (short)0, accS, false, false);
        accQ = __builtin_amdgcn_wmma_f32_16x16x4_f32(false, a2, false, q2,
                                                     (short)0, accQ, false, false);
    }

    // Flush: D layout -> VGPR j, lanes0-15: M=j; lanes16-31: M=j+8; N = lane%16
#pragma unroll
    for (int j = 0; j < 8; ++j) {
        const int m = j + 8 * (int)half;
        atomicAdd(&S[m * CDIM + c], accS[j]);
        float q = accQ[j];
        q += __shfl_xor(q, 1, 32);
        q += __shfl_xor(q, 2, 32);
        q += __shfl_xor(q, 4, 32);
        q += __shfl_xor(q, 8, 32);                  // sum over the 16-lane half
        if (nIdx == 0) atomicAdd(&Qg[m], q);
    }
}

// ---------------------------------------------------------------- kernel 3
__global__ __launch_bounds__(256) void k_final(const float* __restrict__ counts,
                                               const float* __restrict__ Qg,
                                               const float* __restrict__ S,
                                               float* __restrict__ out) {
    __shared__ float ns[KCLS][16];
    __shared__ float pcv[KCLS];
    __shared__ int   vld[KCLS];
    int t = threadIdx.x;                 // 256 threads
    int k = t >> 4, part = t & 15;

    float acc = 0.0f;
    const float* row = S + k * CDIM + part * 16;
#pragma unroll
    for (int j = 0; j < 16; ++j) acc += row[j] * row[j];
    ns[k][part] = acc;
    __syncthreads();

    if (part == 0) {
        float tot = 0.0f;
        for (int j = 0; j < 16; ++j) tot += ns[k][j];
        float cnt = counts[k];
        int v = (k != 0) && (cnt > 0.0f);
        pcv[k] = v ? (Qg[k] - tot / cnt) / cnt : 0.0f;
        vld[k] = v;
    }
    __syncthreads();

    if (t == 0) {
        float s = 0.0f, nv = 0.0f;
        for (int j = 0; j < KCLS; ++j) { s += pcv[j]; nv += (float)vld[j]; }
        out[0] = (nv > 0.0f) ? (s / nv) : 0.0f;   // FACTOR == 1.0
    }
}

// ----------------------------------------------------------------
extern "C" void kernel_launch(void* const* d_in, const int* in_sizes, int n_in,
                              void* d_out, int out_size, void* d_ws, size_t ws_size,
                              hipStream_t stream) {
    const float* f      = (const float*)d_in[0];
    const float* fo     = (const float*)d_in[1];
    const float* oo     = (const float*)d_in[2];
    const int*   labels = (const int*)d_in[3];
    // d_in[4] = prototypes: unused by the reference
    const int*   numold = (const int*)d_in[5];

    const int N = in_sizes[3];           // B*H*W = 262144

    // workspace layout
    float* counts = (float*)d_ws;                      // 16
    float* Qg     = counts + KCLS;                     // 16
    float* S      = Qg + KCLS;                         // 16*256
    unsigned char* seg = (unsigned char*)(S + KCLS * CDIM);  // N bytes

    k_zero<<<1, 256, 0, stream>>>(counts, KCLS + KCLS + KCLS * CDIM);
    k_pseudo<<<N / 256, 256, 0, stream>>>(oo, labels, numold, seg, counts);
    dim3 g2(N / PIXB, 2);
    k_main<<<g2, 256, 0, stream>>>(f, fo, seg, S, Qg);
    k_final<<<1, 256, 0, stream>>>(counts, Qg, S, (float*)d_out);
}